// SVQIModule_82918638616715
// MI455X (gfx1250) — compile-verified
//
#include <hip/hip_runtime.h>
#include <hip/hip_bf16.h>
#include <math.h>

#define B_ 4
#define K_ 256
#define C_ 256
#define L_ 100000
#define N_ 64
#define H_ 128
#define RADIUS2 2.25f
#define CHUNK 1024
#define CAP 1024

typedef float v2f __attribute__((ext_vector_type(2)));
typedef float v8f __attribute__((ext_vector_type(8)));
typedef unsigned int v4u __attribute__((ext_vector_type(4)));
typedef int v4i __attribute__((ext_vector_type(4)));
typedef int v8i __attribute__((ext_vector_type(8)));

#if defined(__gfx1250__) && __has_builtin(__builtin_amdgcn_tensor_load_to_lds) && __has_builtin(__builtin_amdgcn_s_wait_tensorcnt)
#define USE_TDM 1
#else
#define USE_TDM 0
#endif

#if defined(__gfx1250__) && __has_builtin(__builtin_amdgcn_wmma_f32_16x16x4_f32)
#define WMMA_OK 1
#else
#define WMMA_OK 0
#endif

static __device__ __host__ inline int imin(int a, int b) { return a < b ? a : b; }

// ---------------------------------------------------------------------------
// Kernel 1: voxel centers (x,y,z, batch-bits) from sparse indices [b,z,y,x]
// ---------------------------------------------------------------------------
__global__ void vxyz_kernel(const int* __restrict__ sp_idx,
                            float4* __restrict__ vxyz) {
  int l = blockIdx.x * blockDim.x + threadIdx.x;
  if (l >= L_) return;
  int b = sp_idx[l * 4 + 0];
  int z = sp_idx[l * 4 + 1];
  int y = sp_idx[l * 4 + 2];
  int x = sp_idx[l * 4 + 3];
  float4 v;
  v.x = ((float)x + 0.5f) * 0.2f + (-25.6f);
  v.y = ((float)y + 0.5f) * 0.2f + (-25.6f);
  v.z = ((float)z + 0.5f) * 0.2f + (-2.0f);
  v.w = __int_as_float(b);
  vxyz[l] = v;
}

// ---------------------------------------------------------------------------
// TDM issue: 1-D tile of `elems` f32 from global into LDS (descriptor per ISA
// CDNA5 ISA). Issued by one wave; EXEC ignored by TDM.
// 6-arg builtin form (clang-23 / therock-10.0 headers).
// ---------------------------------------------------------------------------
#if USE_TDM
__device__ inline void tdm_issue(const void* gptr, void* lptr, int elems) {
  unsigned long long ga = (unsigned long long)(size_t)gptr;
  unsigned lds = (unsigned)(size_t)lptr;  // flat LDS aperture: low 32 bits = LDS offset
  v4u g0;
  g0[0] = 1u;                                       // count=1, user descriptor
  g0[1] = lds;                                      // lds_addr
  g0[2] = (unsigned)(ga & 0xffffffffu);             // global_addr[31:0]
  g0[3] = (unsigned)((ga >> 32) & 0x1ffffffu) | (2u << 30);  // addr[56:32] | type=2
  v8i g1;
  g1[0] = (int)(2u << 16);                          // data_size=4B
  g1[1] = (int)((unsigned)(elems & 0xffff) << 16);  // tensor_dim0[15:0]
  g1[2] = (int)(((unsigned)(elems >> 16) & 0xffffu) | (1u << 16));  // dim0 hi | tensor_dim1=1
  g1[3] = (int)((unsigned)(elems & 0xffff) << 16);  // tile_dim0 = elems
  g1[4] = 0;                                        // tile_dim1=0 (1-D), tile_dim2=0
  g1[5] = elems;                                    // tensor_dim0_stride lo
  g1[6] = 0;
  g1[7] = 0;
  v4i z4 = {0, 0, 0, 0};
  v8i z8 = {0, 0, 0, 0, 0, 0, 0, 0};
  __builtin_amdgcn_tensor_load_to_lds(g0, g1, z4, z4, z8, 0);
}
#endif

// ---------------------------------------------------------------------------
// Kernel 2: per-query neighbor selection. TDM-streamed voxel scan + radius
// filter into LDS candidate buffer, then 64 rounds of parallel arg-min
// on (d2_bits<<32 | idx) keys (matches top_k tie ordering).
// ---------------------------------------------------------------------------
__global__ void __launch_bounds__(256) select_kernel(
    const float* __restrict__ q_xyz, const float4* __restrict__ vxyz,
    int* __restrict__ nidx_out) {
  extern __shared__ char smem[];
  float4* buf = (float4*)smem;                                    // 2*CHUNK
  unsigned long long* cand = (unsigned long long*)(smem + 2 * CHUNK * 16);  // CAP
  unsigned long long* red = cand + CAP;                           // 256
  int* cnt = (int*)(red + 256);

  const int q = blockIdx.x;
  const int b = q / K_;
  const int tid = threadIdx.x;
  const float qx = q_xyz[q * 3 + 0];
  const float qy = q_xyz[q * 3 + 1];
  const float qz = q_xyz[q * 3 + 2];

  if (tid == 0) *cnt = 0;
  for (int i = tid; i < CAP; i += 256) cand[i] = ~0ULL;
  __syncthreads();

  const int nch = (L_ + CHUNK - 1) / CHUNK;
#if USE_TDM
  if (tid < 32) {
    tdm_issue(vxyz, buf, imin(CHUNK, L_) * 4);
  }
#endif
  for (int c = 0; c < nch; ++c) {
    const int base = c * CHUNK;
    const int rows = imin(CHUNK, L_ - base);
    float4* cur = buf + (size_t)(c & 1) * CHUNK;
#if USE_TDM
    if (tid < 32) {
      if (c + 1 < nch) {
        int nb = (c + 1) * CHUNK;
        tdm_issue(vxyz + nb, buf + (size_t)((c + 1) & 1) * CHUNK,
                  imin(CHUNK, L_ - nb) * 4);
        __builtin_amdgcn_s_wait_tensorcnt(1);  // chunk c complete, c+1 in flight
      } else {
        __builtin_amdgcn_s_wait_tensorcnt(0);
      }
    }
    __syncthreads();
#else
    for (int r = tid; r < rows; r += 256) cur[r] = vxyz[base + r];
    __syncthreads();
#endif
    for (int r = tid; r < rows; r += 256) {
      float4 v = cur[r];
      if (__float_as_int(v.w) == b) {
        float dx = v.x - qx, dy = v.y - qy, dz = v.z - qz;
        float d2 = dx * dx + dy * dy + dz * dz;
        if (d2 <= RADIUS2) {
          int p = atomicAdd(cnt, 1);
          if (p < CAP)
            cand[p] = ((unsigned long long)__float_as_uint(d2) << 32) |
                      (unsigned)(base + r);
        }
      }
    }
    __syncthreads();  // protect buffer before next TDM overwrite
  }

  // 64 rounds of parallel arg-min over candidate keys
  for (int j = 0; j < N_; ++j) {
    unsigned long long loc = ~0ULL;
    for (int i = tid; i < CAP; i += 256) {
      unsigned long long k = cand[i];
      loc = k < loc ? k : loc;
    }
    red[tid] = loc;
    __syncthreads();
    for (int s = 128; s > 0; s >>= 1) {
      if (tid < s) {
        unsigned long long o = red[tid + s];
        if (o < red[tid]) red[tid] = o;
      }
      __syncthreads();
    }
    unsigned long long win = red[0];
    if (win != ~0ULL) {
      for (int i = tid; i < CAP; i += 256)
        if (cand[i] == win) cand[i] = ~0ULL;
      if (tid == 0) nidx_out[q * N_ + j] = (int)(unsigned)(win & 0xffffffffu);
    } else {
      if (tid == 0) nidx_out[q * N_ + j] = -1;
    }
    __syncthreads();
  }
}

// ---------------------------------------------------------------------------
// Kernel 3: gather + pos-MLP (layer2 via V_WMMA_F32_16X16X4_F32) + fused
// single-query attention (K/V/O GEMMs algebraically reduced to matvecs).
// ---------------------------------------------------------------------------
__global__ void __launch_bounds__(256) attn_kernel(
    const float* __restrict__ q_feat, const float* __restrict__ q_xyz,
    const float* __restrict__ sp_feat, const float4* __restrict__ vxyz,
    const int* __restrict__ nidx,
    const float* __restrict__ pos_w1, const float* __restrict__ pos_b1,
    const float* __restrict__ pos_w2, const float* __restrict__ pos_b2,
    const float* __restrict__ wq, const float* __restrict__ bq,
    const float* __restrict__ wk, const float* __restrict__ bk,
    const float* __restrict__ wv, const float* __restrict__ bv,
    const float* __restrict__ wo, const float* __restrict__ bo,
    float* __restrict__ out) {
  extern __shared__ char smem[];
  float* val = (float*)smem;        // 64*256  (vf, then vf+pe)
  float* hbuf = val + N_ * C_;      // 64*128
  float* qv = hbuf + N_ * H_;       // 256 (q, later agg)
  float* qk = qv + C_;              // 256
  float* av = qk + C_;              // 256
  float* red = av + C_;             // 256
  float* ri = red + C_;             // 64*4 rel_in
  float* sc = ri + N_ * 4;          // 64
  float* at = sc + N_;              // 64
  int* nid = (int*)(at + N_);       // 64
  float* sca = (float*)(nid + N_);  // scalars

  const int q = blockIdx.x;
  const int tid = threadIdx.x;
  const float qx = q_xyz[q * 3 + 0];
  const float qy = q_xyz[q * 3 + 1];
  const float qz = q_xyz[q * 3 + 2];

  // neighbor ids + rel_in = [rel_xyz, |rel|]
  if (tid < N_) {
    int i = nidx[q * N_ + tid];
    nid[tid] = i;
    float rx = 0.f, ry = 0.f, rz = 0.f, rd = 0.f;
    if (i >= 0) {
      float4 v = vxyz[i];
      rx = v.x - qx; ry = v.y - qy; rz = v.z - qz;
      rd = sqrtf(rx * rx + ry * ry + rz * rz);
    }
    ri[tid * 4 + 0] = rx; ri[tid * 4 + 1] = ry;
    ri[tid * 4 + 2] = rz; ri[tid * 4 + 3] = rd;
  }
  __syncthreads();

  // gather voxel features (float4) into val
  for (int e = tid; e < N_ * C_ / 4; e += 256) {
    int n = e >> 6;     // 64 float4 per row
    int c4 = e & 63;
    int i = nid[n];
    float4 v = make_float4(0.f, 0.f, 0.f, 0.f);
    if (i >= 0) v = ((const float4*)sp_feat)[(size_t)i * 64 + c4];
    ((float4*)val)[n * 64 + c4] = v;
  }
  // pos-MLP layer 1: h = relu(rel_in @ w1.T + b1), 64x128
  for (int e = tid; e < N_ * H_; e += 256) {
    int n = e >> 7;
    int j = e & 127;
    float s = pos_b1[j] + ri[n * 4 + 0] * pos_w1[j * 4 + 0] +
              ri[n * 4 + 1] * pos_w1[j * 4 + 1] +
              ri[n * 4 + 2] * pos_w1[j * 4 + 2] +
              ri[n * 4 + 3] * pos_w1[j * 4 + 3];
    hbuf[e] = s > 0.f ? s : 0.f;
  }
  __syncthreads();

  // pos-MLP layer 2 via fp32 WMMA: val += h(64x128) @ pos_w2^T(128x256) + b2
  {
    const int wave = tid >> 5;
    const int lane = tid & 31;
    const int lm = lane & 15;
    const int lk = (lane >> 4) << 1;  // K sub-offset 0 or 2
    for (int t = 0; t < 8; ++t) {
      const int tile = wave * 8 + t;   // 64 tiles: 4 (M) x 16 (N)
      const int mT = tile >> 4, nT = tile & 15;
      const int m = mT * 16 + lm;
      const int n = nT * 16 + lm;
#if WMMA_OK
      v8f acc = {};
#pragma unroll
      for (int k0 = 0; k0 < H_; k0 += 4) {
        v2f a, bb;
        a.x = hbuf[m * H_ + k0 + lk];
        a.y = hbuf[m * H_ + k0 + lk + 1];
        bb.x = pos_w2[n * H_ + k0 + lk];      // B[k][n] = pos_w2[n][k]
        bb.y = pos_w2[n * H_ + k0 + lk + 1];
        acc = __builtin_amdgcn_wmma_f32_16x16x4_f32(
            false, a, false, bb, (short)0, acc, false, false);
      }
#pragma unroll
      for (int r = 0; r < 8; ++r) {
        int mr = mT * 16 + r + ((lane >> 4) << 3);
        val[mr * C_ + n] += acc[r] + pos_b2[n];
      }
#else
      for (int r = 0; r < 8; ++r) {
        int mr = mT * 16 + r + ((lane >> 4) << 3);
        float s = 0.f;
        for (int k0 = 0; k0 < H_; ++k0) s += hbuf[mr * H_ + k0] * pos_w2[n * H_ + k0];
        val[mr * C_ + n] += s + pos_b2[n];
      }
#endif
    }
  }
  __syncthreads();

  // q = q_feat @ wq.T + bq   (one output channel per thread)
  {
    const float* qf = q_feat + (size_t)q * C_;
    const float* w = wq + (size_t)tid * C_;
    __builtin_prefetch(wk + (size_t)tid * C_, 0, 0);
    float s = bq[tid];
    for (int c = 0; c < C_; ++c) s += qf[c] * w[c];
    qv[tid] = s;
  }
  __syncthreads();
  // qk = q @ wk (coalesced over columns); red = q*bk partials
  {
    float s = 0.f;
    for (int c = 0; c < C_; ++c) s += qv[c] * wk[(size_t)c * C_ + tid];
    qk[tid] = s;
    red[tid] = qv[tid] * bk[tid];
  }
  __syncthreads();
  for (int s = 128; s > 0; s >>= 1) {
    if (tid < s) red[tid] += red[tid + s];
    __syncthreads();
  }
  const float qbk = red[0];

  // scores_n = (val_n . qk + q.bk) / sqrt(C), masked
  if (tid < N_) {
    const float* vr = val + tid * C_;
    float s = qbk;
    for (int c = 0; c < C_; ++c) s += vr[c] * qk[c];
    s *= 0.0625f;  // 1/sqrt(256)
    sc[tid] = (nid[tid] >= 0) ? s : -1.0e9f;
  }
  __syncthreads();
  if (tid == 0) {
    float mx = -INFINITY;
    for (int n = 0; n < N_; ++n) mx = fmaxf(mx, sc[n]);
    float sum = 0.f;
    for (int n = 0; n < N_; ++n) { float e = expf(sc[n] - mx); at[n] = e; sum += e; }
    float inv = 1.0f / sum;
    for (int n = 0; n < N_; ++n) at[n] *= inv;
    float anyv = 0.f;
    for (int n = 0; n < N_; ++n) if (nid[n] >= 0) anyv = 1.f;
    sca[0] = anyv;
  }
  __syncthreads();
  const float anyv = sca[0];

  // aval = attn @ val  (256-vec)
  {
    float s = 0.f;
    for (int n = 0; n < N_; ++n) s += at[n] * val[n * C_ + tid];
    av[tid] = s;
  }
  __syncthreads();
  // agg = aval @ wv.T + bv
  {
    const float* w = wv + (size_t)tid * C_;
    float s = bv[tid];
    for (int c = 0; c < C_; ++c) s += av[c] * w[c];
    qv[tid] = s;
  }
  __syncthreads();
  // out = (agg @ wo.T + bo) * any_valid
  {
    const float* w = wo + (size_t)tid * C_;
    float s = bo[tid];
    for (int c = 0; c < C_; ++c) s += qv[c] * w[c];
    out[(size_t)q * C_ + tid] = anyv * s;
  }
}

// ---------------------------------------------------------------------------
extern "C" void kernel_launch(void* const* d_in, const int* in_sizes, int n_in,
                              void* d_out, int out_size, void* d_ws,
                              size_t ws_size, hipStream_t stream) {
  const float* q_feat = (const float*)d_in[0];
  const float* q_xyz = (const float*)d_in[1];
  const float* sp_feat = (const float*)d_in[2];
  const int* sp_idx = (const int*)d_in[3];
  const float* pos_w1 = (const float*)d_in[4];
  const float* pos_b1 = (const float*)d_in[5];
  const float* pos_w2 = (const float*)d_in[6];
  const float* pos_b2 = (const float*)d_in[7];
  const float* wq = (const float*)d_in[8];
  const float* bq = (const float*)d_in[9];
  const float* wk = (const float*)d_in[10];
  const float* bk = (const float*)d_in[11];
  const float* wv = (const float*)d_in[12];
  const float* bv = (const float*)d_in[13];
  const float* wo = (const float*)d_in[14];
  const float* bo = (const float*)d_in[15];
  float* out = (float*)d_out;

  float4* vxyz = (float4*)d_ws;
  int* nidx = (int*)((char*)d_ws + (size_t)L_ * sizeof(float4));

  vxyz_kernel<<<(L_ + 255) / 256, 256, 0, stream>>>(sp_idx, vxyz);

  size_t sm2 = (size_t)2 * CHUNK * sizeof(float4) + (size_t)CAP * 8 + 256 * 8 + 16;
  select_kernel<<<B_ * K_, 256, sm2, stream>>>(q_xyz, vxyz, nidx);

  size_t sm3 = (size_t)(N_ * C_ + N_ * H_ + 4 * C_ + N_ * 4 + 2 * N_) * sizeof(float) +
               N_ * sizeof(int) + 64;
  attn_kernel<<<B_ * K_, 256, sm3, stream>>>(q_feat, q_xyz, sp_feat, vxyz, nidx,
                                             pos_w1, pos_b1, pos_w2, pos_b2,
                                             wq, bq, wk, bk, wv, bv, wo, bo, out);
}